// MolecularGnn_32186484916934
// MI455X (gfx1250) — compile-verified
//
#include <hip/hip_runtime.h>
#include <hip/hip_bf16.h>
#include <math.h>

typedef __attribute__((ext_vector_type(2))) float v2f;
typedef __attribute__((ext_vector_type(8))) float v8f;

#define N_NODES 50000
#define N_EDGES 800000
#define N_ATOMS 100
#define HID     128
#define N_GRAPHS 1024

// ---------------------------------------------------------------------------
// Zero a float buffer
// ---------------------------------------------------------------------------
__global__ __launch_bounds__(256)
void zero_f(float* __restrict__ p, int n) {
    int i = blockIdx.x * 256 + threadIdx.x;
    if (i < n) p[i] = 0.0f;
}

// ---------------------------------------------------------------------------
// X[n][:] = embd_w[x_ori[n]][:]   (one wave32 per node, float4 per lane)
// ---------------------------------------------------------------------------
__global__ __launch_bounds__(256)
void gather_embed(const int* __restrict__ x_ori, const float* __restrict__ embd,
                  float* __restrict__ X, int nnodes) {
    int n = blockIdx.x * 8 + (threadIdx.x >> 5);
    if (n >= nnodes) return;
    int lane = threadIdx.x & 31;
    int atom = x_ori[n];
    float4 v = *(const float4*)(embd + (size_t)atom * HID + lane * 4);
    *(float4*)(X + (size_t)n * HID + lane * 4) = v;
}

// ---------------------------------------------------------------------------
// Y = relu(X @ W + b), X:[nrows,128], W:[128,128] row-major, b:[128]
// fp32 WMMA 16x16x4. 8 waves/block -> 128 rows x 128 cols per block.
// Fast-path epilogue: wave-uniform bounds check so 390/391 blocks store
// without any EXEC manipulation.
// ---------------------------------------------------------------------------
__global__ __launch_bounds__(256)
void gemm_relu_wmma(const float* __restrict__ X, const float* __restrict__ W,
                    const float* __restrict__ bias, float* __restrict__ Y,
                    int nrows) {
    __shared__ float sW[HID * HID];  // 64 KB, staged once per block
    for (int i = threadIdx.x; i < HID * HID; i += 256) sW[i] = W[i];
    __syncthreads();

    const int lane  = threadIdx.x & 31;
    const int wave  = threadIdx.x >> 5;
    const int row0  = blockIdx.x * 128 + wave * 16;  // 16 rows per wave
    const int m     = lane & 15;
    const int khalf = (lane >> 4) * 2;               // 0 for lanes 0-15, 2 for 16-31

    v8f acc[8];
#pragma unroll
    for (int t = 0; t < 8; ++t)
#pragma unroll
        for (int r = 0; r < 8; ++r) acc[t][r] = 0.0f;

    const int  arow   = row0 + m;
    const int  safrow = (arow < nrows) ? arow : (nrows - 1);  // clamp (EXEC stays all-1s)
    const float* Arow = X + (size_t)safrow * HID;

    for (int k0 = 0; k0 < HID; k0 += 4) {
        v2f a;
        a.x = Arow[k0 + khalf];
        a.y = Arow[k0 + khalf + 1];
#pragma unroll
        for (int t = 0; t < 8; ++t) {
            const int n = t * 16 + m;
            v2f b;
            b.x = sW[(k0 + khalf) * HID + n];
            b.y = sW[(k0 + khalf + 1) * HID + n];
            acc[t] = __builtin_amdgcn_wmma_f32_16x16x4_f32(
                false, a, false, b, (short)0, acc[t], false, false);
        }
    }

    // Bias for the 8 column tiles this lane touches (col = t*16 + m)
    float bv[8];
#pragma unroll
    for (int t = 0; t < 8; ++t) bv[t] = bias[t * 16 + m];

    // C/D layout: vgpr r, lanes 0-15 -> M=r ; lanes 16-31 -> M=r+8 ; N = lane%16
    const int rofs = (lane >> 4) * 8;
    float* Yw = Y + (size_t)(row0 + rofs) * HID + m;

    if (row0 + 16 <= nrows) {
        // Full tile: straight-line unguarded stores (common case)
#pragma unroll
        for (int r = 0; r < 8; ++r) {
            float* Yr = Yw + (size_t)r * HID;
#pragma unroll
            for (int t = 0; t < 8; ++t) {
                Yr[t * 16] = fmaxf(acc[t][r] + bv[t], 0.0f);
            }
        }
    } else {
        // Tail block: per-row guard
#pragma unroll
        for (int r = 0; r < 8; ++r) {
            const int row = row0 + r + rofs;
            if (row < nrows) {
                float* Yr = Yw + (size_t)r * HID;
#pragma unroll
                for (int t = 0; t < 8; ++t) {
                    Yr[t * 16] = fmaxf(acc[t][r] + bv[t], 0.0f);
                }
            }
        }
    }
}

// ---------------------------------------------------------------------------
// Edge scatter: agg[dst] += exp(-sigmoid(gamma[atom[dst]])*ea^2) * h[src]
// One wave32 per edge; float4 per lane; native f32 global atomics (L2-resident).
// ---------------------------------------------------------------------------
__global__ __launch_bounds__(256)
void edge_scatter(const int* __restrict__ edge_index, const float* __restrict__ edge_attr,
                  const int* __restrict__ x_ori, const float* __restrict__ gamma_l,
                  const float* __restrict__ Hbuf, float* __restrict__ AGG, int nedges) {
    int e = blockIdx.x * 8 + (threadIdx.x >> 5);
    if (e >= nedges) return;
    int lane = threadIdx.x & 31;

    int src = edge_index[e];
    int dst = edge_index[nedges + e];
    float g   = gamma_l[x_ori[dst]];
    float sg  = 1.0f / (1.0f + expf(-g));
    float ea  = edge_attr[e];
    float w   = expf(-sg * ea * ea);

    float4 h4 = *(const float4*)(Hbuf + (size_t)src * HID + lane * 4);
    float* p  = AGG + (size_t)dst * HID + lane * 4;
    __hip_atomic_fetch_add(p + 0, w * h4.x, __ATOMIC_RELAXED, __HIP_MEMORY_SCOPE_AGENT);
    __hip_atomic_fetch_add(p + 1, w * h4.y, __ATOMIC_RELAXED, __HIP_MEMORY_SCOPE_AGENT);
    __hip_atomic_fetch_add(p + 2, w * h4.z, __ATOMIC_RELAXED, __HIP_MEMORY_SCOPE_AGENT);
    __hip_atomic_fetch_add(p + 3, w * h4.w, __ATOMIC_RELAXED, __HIP_MEMORY_SCOPE_AGENT);
}

// ---------------------------------------------------------------------------
// X = normalize(AGG + X) rowwise (L2, eps=1e-12). One wave32 per node.
// ---------------------------------------------------------------------------
__global__ __launch_bounds__(256)
void add_normalize(const float* __restrict__ AGG, float* __restrict__ X, int nnodes) {
    int n = blockIdx.x * 8 + (threadIdx.x >> 5);
    if (n >= nnodes) return;
    int lane = threadIdx.x & 31;
    size_t base = (size_t)n * HID + lane * 4;
    float4 a = *(const float4*)(AGG + base);
    float4 x = *(const float4*)(X + base);
    float4 y;
    y.x = a.x + x.x; y.y = a.y + x.y; y.z = a.z + x.z; y.w = a.w + x.w;
    float ss = y.x * y.x + y.y * y.y + y.z * y.z + y.w * y.w;
#pragma unroll
    for (int off = 16; off > 0; off >>= 1) ss += __shfl_xor(ss, off, 32);
    float inv = 1.0f / fmaxf(sqrtf(ss), 1e-12f);
    y.x *= inv; y.y *= inv; y.z *= inv; y.w *= inv;
    *(float4*)(X + base) = y;
}

// ---------------------------------------------------------------------------
// Graph pooling: MX[batch[n]] += X[n]. One wave32 per node.
// ---------------------------------------------------------------------------
__global__ __launch_bounds__(256)
void graph_pool(const float* __restrict__ X, const int* __restrict__ batch_ids,
                float* __restrict__ MX, int nnodes) {
    int n = blockIdx.x * 8 + (threadIdx.x >> 5);
    if (n >= nnodes) return;
    int lane = threadIdx.x & 31;
    int g = batch_ids[n];
    float4 v = *(const float4*)(X + (size_t)n * HID + lane * 4);
    float* p = MX + (size_t)g * HID + lane * 4;
    __hip_atomic_fetch_add(p + 0, v.x, __ATOMIC_RELAXED, __HIP_MEMORY_SCOPE_AGENT);
    __hip_atomic_fetch_add(p + 1, v.y, __ATOMIC_RELAXED, __HIP_MEMORY_SCOPE_AGENT);
    __hip_atomic_fetch_add(p + 2, v.z, __ATOMIC_RELAXED, __HIP_MEMORY_SCOPE_AGENT);
    __hip_atomic_fetch_add(p + 3, v.w, __ATOMIC_RELAXED, __HIP_MEMORY_SCOPE_AGENT);
}

// ---------------------------------------------------------------------------
// props[g] = dot(MX[g], wp_W) + wp_b. One wave32 per graph.
// ---------------------------------------------------------------------------
__global__ __launch_bounds__(256)
void props_kernel(const float* __restrict__ MX, const float* __restrict__ wpW,
                  const float* __restrict__ wpb, float* __restrict__ out, int ngraphs) {
    int g = blockIdx.x * 8 + (threadIdx.x >> 5);
    if (g >= ngraphs) return;
    int lane = threadIdx.x & 31;
    float4 m = *(const float4*)(MX + (size_t)g * HID + lane * 4);
    float4 w = *(const float4*)(wpW + lane * 4);
    float s = m.x * w.x + m.y * w.y + m.z * w.z + m.w * w.w;
#pragma unroll
    for (int off = 16; off > 0; off >>= 1) s += __shfl_xor(s, off, 32);
    if (lane == 0) out[g] = s + wpb[0];
}

// ---------------------------------------------------------------------------
extern "C" void kernel_launch(void* const* d_in, const int* in_sizes, int n_in,
                              void* d_out, int out_size, void* d_ws, size_t ws_size,
                              hipStream_t stream) {
    const int*   x_ori      = (const int*)d_in[0];
    const int*   edge_index = (const int*)d_in[1];
    const float* edge_attr  = (const float*)d_in[2];
    const int*   batch_ids  = (const int*)d_in[3];
    const float* embd_w     = (const float*)d_in[4];
    const float* gamma_w    = (const float*)d_in[5];   // [3,100,1]
    const float* w_atom_W   = (const float*)d_in[6];   // [3,128,128]
    const float* w_atom_b   = (const float*)d_in[7];   // [3,128]
    const float* lin_W      = (const float*)d_in[8];   // [2,128,128]
    const float* lin_b      = (const float*)d_in[9];   // [2,128]
    const float* wp_W       = (const float*)d_in[10];  // [128,1]
    const float* wp_b       = (const float*)d_in[11];  // [1]
    float* out = (float*)d_out;

    const int N = N_NODES, E = N_EDGES, G = N_GRAPHS;
    const size_t NH = (size_t)N * HID;

    float* X   = (float*)d_ws;
    float* Hb  = X + NH;
    float* AGG = Hb + NH;
    float* MX  = AGG + NH;

    const dim3 blk(256);
    const int nodeBlocks = (N + 7) / 8;          // wave-per-node kernels
    const int edgeBlocks = (E + 7) / 8;          // wave-per-edge
    const int gemmBlocks = (N + 127) / 128;      // 128 rows per block
    const int zeroNH     = ((int)NH + 255) / 256;
    const int zeroMX     = (G * HID + 255) / 256;
    const int graphBlks  = (G + 7) / 8;

    // x = embd_w[x_ori]
    gather_embed<<<nodeBlocks, blk, 0, stream>>>(x_ori, embd_w, X, N);

    for (int l = 0; l < 3; ++l) {
        // h = relu(x @ W_l + b_l)
        gemm_relu_wmma<<<gemmBlocks, blk, 0, stream>>>(
            X, w_atom_W + (size_t)l * HID * HID, w_atom_b + (size_t)l * HID, Hb, N);
        // agg = 0
        zero_f<<<zeroNH, blk, 0, stream>>>(AGG, (int)NH);
        // agg[dst] += exp(-sigmoid(gamma)*ea^2) * h[src]
        edge_scatter<<<edgeBlocks, blk, 0, stream>>>(
            edge_index, edge_attr, x_ori, gamma_w + (size_t)l * N_ATOMS, Hb, AGG, E);
        // x = normalize(agg + x)
        add_normalize<<<nodeBlocks, blk, 0, stream>>>(AGG, X, N);
    }

    // out MLP: X -> Hb -> X  (both relu)
    gemm_relu_wmma<<<gemmBlocks, blk, 0, stream>>>(X, lin_W, lin_b, Hb, N);
    gemm_relu_wmma<<<gemmBlocks, blk, 0, stream>>>(
        Hb, lin_W + (size_t)HID * HID, lin_b + HID, X, N);

    // graph pooling + readout
    zero_f<<<zeroMX, blk, 0, stream>>>(MX, G * HID);
    graph_pool<<<nodeBlocks, blk, 0, stream>>>(X, batch_ids, MX, N);
    props_kernel<<<graphBlks, blk, 0, stream>>>(MX, wp_W, wp_b, out, G);
}